// RelativeAttention_20194936226207
// MI455X (gfx1250) — compile-verified
//
#include <hip/hip_runtime.h>
#include <hip/hip_bf16.h>

typedef float v2f __attribute__((ext_vector_type(2)));
typedef float v4f __attribute__((ext_vector_type(4)));
typedef float v8f __attribute__((ext_vector_type(8)));

static constexpr int kBN   = 64;    // B*N batched pairs
static constexpr int kL    = 1024;  // H*W
static constexpr int kD    = 64;    // feature dim
static constexpr int kMREL = 63;    // 2*32-1 relative rows

__device__ __forceinline__ v8f wmma_f32_k4(v2f a, v2f b, v8f c) {
    // D = A(16x4,f32) * B(4x16,f32) + C(16x16,f32)
    return __builtin_amdgcn_wmma_f32_16x16x4_f32(false, a, false, b, (short)0, c, false, false);
}

// ---------------------------------------------------------------------------
// Kernel 1: Qw[bn][q][m] = sum_d Q[bn,q,d] * rel_emb_w[head,m,d]   (m padded to 64)
//           Qh likewise.  One wave computes a 16(q) x 64(m) strip for both tables.
// ---------------------------------------------------------------------------
__global__ __launch_bounds__(128) void rel_tables_kernel(
    const float* __restrict__ q,
    const float* __restrict__ emb_w,
    const float* __restrict__ emb_h,
    float* __restrict__ qw_tbl,
    float* __restrict__ qh_tbl)
{
    const int bn   = blockIdx.x;
    const int head = bn & 7;
    const int wave = threadIdx.x >> 5;
    const int lane = threadIdx.x & 31;
    const int half = lane >> 4;          // 0: K-slots {0,1}, 1: K-slots {2,3}
    const int l16  = lane & 15;
    const int qbase = blockIdx.y * 64 + wave * 16;

    // A panel: 16 query rows, 64 features, loaded as 8 x b128 per lane
    const float* aptr = q + (size_t)(bn * kL + qbase + l16) * kD + 4 * half;
    v4f apan[8];
#pragma unroll
    for (int t = 0; t < 8; ++t) apan[t] = *(const v4f*)(aptr + 8 * t);

#pragma unroll
    for (int which = 0; which < 2; ++which) {
        const float* emb = which ? emb_h : emb_w;
        float* tbl       = which ? qh_tbl : qw_tbl;
#pragma unroll
        for (int nt = 0; nt < 4; ++nt) {
            const int n0 = nt * 16;
            int m = n0 + l16;            // embedding row; clamp pad column 63
            if (m > kMREL - 1) m = kMREL - 1;
            const float* bptr = emb + (size_t)(head * kMREL + m) * kD + 4 * half;
            v4f bpan[8];
#pragma unroll
            for (int t = 0; t < 8; ++t) bpan[t] = *(const v4f*)(bptr + 8 * t);

            v8f acc = {};
#pragma unroll
            for (int t = 0; t < 8; ++t) {
                acc = wmma_f32_k4(apan[t].xy, bpan[t].xy, acc);
                acc = wmma_f32_k4(apan[t].zw, bpan[t].zw, acc);
            }
#pragma unroll
            for (int v = 0; v < 8; ++v) {
                const int row = qbase + v + 8 * half;
                tbl[(size_t)(bn * kL + row) * kD + n0 + l16] = acc[v];
            }
        }
    }
}

// ---------------------------------------------------------------------------
// Kernel 2: logits[bn][q][key] = Q·K^T + Qw[q, y-w+31] + Qh[q, x-h+31]
// Block: 256 threads = 8 waves, arranged 4(M) x 2(N): 64 queries x 128 keys.
// Each wave: 16 queries x 64 keys (4 WMMA tiles), A panel register-resident.
// ---------------------------------------------------------------------------
__global__ __launch_bounds__(256) void rel_attn_main_kernel(
    const float* __restrict__ q,
    const float* __restrict__ k,
    const float* __restrict__ qw_tbl,
    const float* __restrict__ qh_tbl,
    float* __restrict__ out)
{
    const int bn    = blockIdx.x;
    const int wave  = threadIdx.x >> 5;
    const int lane  = threadIdx.x & 31;
    const int half  = lane >> 4;
    const int l16   = lane & 15;
    const int mwave = wave & 3;
    const int nwave = wave >> 2;
    const int qbase  = blockIdx.y * 64 + mwave * 16;
    const int kgbase = blockIdx.z * 128 + nwave * 64;

    // A panel: 16 query rows x 64 features (8 x b128 per lane)
    const float* aptr = q + (size_t)(bn * kL + qbase + l16) * kD + 4 * half;
    v4f apan[8];
#pragma unroll
    for (int t = 0; t < 8; ++t) apan[t] = *(const v4f*)(aptr + 8 * t);

    const int hh = qbase >> 5;           // query row h, constant across the 16-row tile

#pragma unroll
    for (int kt = 0; kt < 4; ++kt) {
        const int kb = kgbase + kt * 16;

        // B panel: 16 key rows x 64 features, same per-lane quad pattern as A
        const float* bptr = k + (size_t)(bn * kL + kb + l16) * kD + 4 * half;
        v4f bpan[8];
#pragma unroll
        for (int t = 0; t < 8; ++t) bpan[t] = *(const v4f*)(bptr + 8 * t);

        v8f acc = {};
#pragma unroll
        for (int t = 0; t < 8; ++t) {
            acc = wmma_f32_k4(apan[t].xy, bpan[t].xy, acc);
            acc = wmma_f32_k4(apan[t].zw, bpan[t].zw, acc);
        }

        // bias + store: C/D layout -> lane l, VGPR v holds M = v + 8*(l>=16), N = l&15
        const int xx   = kb >> 5;                 // key row x, constant in tile
        const int yy   = (kb & 31) + l16;         // key col y, per lane
        const int hIdx = xx - hh + 31;            // height-relative index, constant in tile
#pragma unroll
        for (int v = 0; v < 8; ++v) {
            const int qrow = qbase + v + 8 * half;
            const int ww   = qrow & 31;
            const size_t qoff = (size_t)(bn * kL + qrow);
            const float bw = qw_tbl[qoff * kD + (yy - ww + 31)];
            const float bh = qh_tbl[qoff * kD + hIdx];
            out[qoff * (size_t)kL + kb + l16] = acc[v] + bw + bh;
        }
    }
}

extern "C" void kernel_launch(void* const* d_in, const int* in_sizes, int n_in,
                              void* d_out, int out_size, void* d_ws, size_t ws_size,
                              hipStream_t stream) {
    const float* q  = (const float*)d_in[0];  // [8,8,32,32,64]
    const float* k  = (const float*)d_in[1];  // [8,8,32,32,64]
    const float* ew = (const float*)d_in[2];  // [8,63,64]
    const float* eh = (const float*)d_in[3];  // [8,63,64]
    float* out = (float*)d_out;               // [64,1024,1024]

    float* qw_tbl = (float*)d_ws;                          // 64*1024*64 f32 = 16 MB
    float* qh_tbl = qw_tbl + (size_t)kBN * kL * kD;        // another 16 MB

    rel_tables_kernel<<<dim3(kBN, kL / 64), 128, 0, stream>>>(q, ew, eh, qw_tbl, qh_tbl);
    rel_attn_main_kernel<<<dim3(kBN, kL / 64, kL / 128), 256, 0, stream>>>(
        q, k, qw_tbl, qh_tbl, out);
}